// CapsuleRouting_17686675325395
// MI455X (gfx1250) — compile-verified
//
#include <hip/hip_runtime.h>
#include <hip/hip_bf16.h>

// Capsule routing (3 iterations), restructured:
//  * r0 = a/C is constant over C -> iter-0 softmax is exactly 1/16 and `a`
//    never affects any output (softmax over C is shift-invariant in r0).
//  * Accumulated logits = sum_p u * (v1+..+vk) (linear in v) -> keep only w = sum v.
//  * Routing is independent per (batch, spatial) position.
// Thread mapping: thread = (C, s); the full P2=16 pose vector lives in VGPRs:
//  - softmax over C via wave32 shfl_xor (C = lanes 0..15), zero LDS round trips
//  - squash + w state fully thread-local (w in 16 VGPRs)
// B-reduction split across G blocks; deterministic partials in d_ws (no atomics);
// small squash kernel between passes (launch boundary = global sync).
// CDNA5 path: TDM tensor_load_to_lds stages the strided (256 rows x 64B,
// row stride 576B) tile into LDS with 1-dword pad per 16 (stride-17 rows),
// double-buffered on TENSORcnt.
// PASS/G are template params so all index math is constant (no s_div sequences).

#define NB 144                // input capsules
#define NC 16                 // output capsules
#define NP 16                 // pose dim
#define SS 144                // spatial = 12*12
#define TS 16                 // spatial tile per block
#define UST 17                // LDS row stride (16 data + 1 TDM pad dword)
#define STILES (SS / TS)      // 9
#define ROWS (NC * NP)        // 256
#define USTRIDE (ROWS * SS)   // elements per (b,B)
#define PSZ (8 * NC * NP * SS) // 294912 elements per accumulator image

typedef unsigned int v4u __attribute__((ext_vector_type(4)));
typedef int          v8i __attribute__((ext_vector_type(8)));
typedef int          v4i __attribute__((ext_vector_type(4)));

// TDM 2D tile load: 256 rows x 16 floats, row stride 144 floats, into LDS at
// ldsoff with 1 pad dword after every 16 dwords (rows land at stride 17).
__device__ __forceinline__ void tdm_load_tile(const float* gsrc, unsigned ldsoff) {
  unsigned long long ga = (unsigned long long)(uintptr_t)gsrc;
  v4u g0;
  g0[0] = 1u;                                   // count=1 (valid descriptor)
  g0[1] = ldsoff;                               // lds_addr (bytes)
  g0[2] = (unsigned)(ga & 0xFFFFFFFFull);       // global_addr[31:0]
  g0[3] = (unsigned)((ga >> 32) & 0x1FFFFFFull) // global_addr[56:32]
        | (2u << 30);                           // type=2 ("image")
  v8i g1;
  g1[0] = (int)((2u << 16)     // data_size = 4B
              | (1u << 20)     // pad_enable
              | (3u << 22)     // pad_interval: every 16 dwords (64B)
              | (0u << 25));   // pad_amount: 1 dword
  g1[1] = (int)((unsigned)TS << 16);    // tensor_dim0 lo16 = 16
  g1[2] = (int)((unsigned)ROWS << 16);  // tensor_dim1 lo16 = 256
  g1[3] = (int)((unsigned)TS << 16);    // tile_dim0 = 16
  g1[4] = (int)ROWS;                    // tile_dim1 = 256 (tile_dim2 = 0)
  g1[5] = (int)SS;                      // tensor_dim0_stride = 144
  g1[6] = 0;
  g1[7] = 0;
  v4i zz4 = {0, 0, 0, 0};               // 2D tensor: groups 2/3 unused
  v8i zz8 = {0, 0, 0, 0, 0, 0, 0, 0};   // extra group (clang-23 6-arg form)
  __builtin_amdgcn_tensor_load_to_lds(g0, g1, zz4, zz4, zz8, 0);
}

// Accumulation pass: block = (b, s-tile, B-group g); computes the partial
// sum over its B range of c[B,C,s] * u[B,C,p,s] and stores it to part[g].
// PASS==0: c = 1/16 (folded into the squash kernel's pre-scale).
// PASS>0 : c = softmax_C( sum_p u * w ), w read from warr.
template <int PASS, int G>
__global__ void __launch_bounds__(256)
caps_acc_kernel(const float* __restrict__ u, const float* __restrict__ warr,
                float* __restrict__ part)
{
  __shared__ float ubuf[2][ROWS * UST];

  const int tid  = threadIdx.x;
  const int lane = tid & 31;
  const int wid  = tid >> 5;
  const int C  = lane & 15;
  const int sh = lane >> 4;
  const int sl = 2 * wid + sh;       // 0..15 within the s-tile

  int idx = blockIdx.x;
  const int bb = idx / (STILES * G); // constant divisors -> mul/shift
  idx -= bb * (STILES * G);
  const int st = idx / G;
  const int g  = idx - st * G;

  const int s0 = st * TS;
  const int s  = s0 + sl;
  constexpr int Bcnt = NB / G;
  const int B0 = g * Bcnt;

  const float* ub = u + ((size_t)bb * NB + B0) * USTRIDE + s0;

  float wreg[NP];
  if (PASS != 0) {
#pragma unroll
    for (int p = 0; p < NP; ++p)
      wreg[p] = warr[(((size_t)bb * NC + C) * NP + p) * SS + s];
  }

  float sacc[NP];
#pragma unroll
  for (int p = 0; p < NP; ++p) sacc[p] = 0.f;

  if (tid < 32) tdm_load_tile(ub, (unsigned)(uintptr_t)&ubuf[0][0]);

  for (int Bi = 0; Bi < Bcnt; ++Bi) {
    const int cur = Bi & 1;
    if (tid < 32) {
      if (Bi + 1 < Bcnt) {
        tdm_load_tile(ub + (size_t)(Bi + 1) * USTRIDE,
                      (unsigned)(uintptr_t)&ubuf[cur ^ 1][0]);
        __builtin_amdgcn_s_wait_tensorcnt(1); // tile Bi complete (in-order)
      } else {
        __builtin_amdgcn_s_wait_tensorcnt(0);
      }
    }
    __syncthreads(); // tile Bi visible to all waves

    float uv[NP];
#pragma unroll
    for (int p = 0; p < NP; ++p)
      uv[p] = ubuf[cur][(C * NP + p) * UST + sl];

    if (PASS == 0) {
#pragma unroll
      for (int p = 0; p < NP; ++p) sacc[p] += uv[p];
    } else {
      float L = 0.f;
#pragma unroll
      for (int p = 0; p < NP; ++p) L += uv[p] * wreg[p];
      // softmax over C: C occupies lanes 0..15 / 16..31 -> xor over bits 0..3
      float m = L;
#pragma unroll
      for (int o = 1; o < 16; o <<= 1) m = fmaxf(m, __shfl_xor(m, o, 32));
      float e = __expf(L - m);
      float su = e;
#pragma unroll
      for (int o = 1; o < 16; o <<= 1) su += __shfl_xor(su, o, 32);
      const float c = e / su;
#pragma unroll
      for (int p = 0; p < NP; ++p) sacc[p] += c * uv[p];
    }
    __syncthreads(); // done reading ubuf[cur] before it is refilled
  }

#pragma unroll
  for (int p = 0; p < NP; ++p)
    part[(size_t)g * PSZ + (((size_t)bb * NC + C) * NP + p) * SS + s] = sacc[p];
}

// Squash pass: sums the G partials (fixed order -> deterministic), applies
// squash, and updates w (passes 0,1) or writes the outputs (pass 2).
template <int PASS, int G>
__global__ void __launch_bounds__(256)
caps_squash_kernel(const float* __restrict__ part, float* __restrict__ warr,
                   float* __restrict__ vout, float* __restrict__ aout)
{
  const int tid  = threadIdx.x;
  const int lane = tid & 31;
  const int wid  = tid >> 5;
  const int C  = lane & 15;
  const int sl = 2 * wid + (lane >> 4);

  const int bb = blockIdx.x / STILES;
  const int st = blockIdx.x - bb * STILES;
  const int s  = st * TS + sl;

  constexpr float pre = (PASS == 0) ? (1.f / 16.f) : 1.f;

  float v[NP];
  float sn = 0.f;
#pragma unroll
  for (int p = 0; p < NP; ++p) {
    const size_t idx = (((size_t)bb * NC + C) * NP + p) * SS + s;
    float acc = 0.f;
#pragma unroll
    for (int g = 0; g < G; ++g) acc += part[(size_t)g * PSZ + idx];
    const float t = acc * pre;
    v[p] = t;
    sn += t * t;
  }
  const float scale = sn / ((1.f + sn) * sqrtf(sn));

  if (PASS == 0) {
#pragma unroll
    for (int p = 0; p < NP; ++p)
      warr[(((size_t)bb * NC + C) * NP + p) * SS + s] = v[p] * scale;
  } else if (PASS == 1) {
#pragma unroll
    for (int p = 0; p < NP; ++p)
      warr[(((size_t)bb * NC + C) * NP + p) * SS + s] += v[p] * scale;
  } else {
    float sn2 = 0.f;
#pragma unroll
    for (int p = 0; p < NP; ++p) {
      const float vv = v[p] * scale;
      vout[(((size_t)bb * NC + C) * NP + p) * SS + s] = vv;
      sn2 += vv * vv;
    }
    aout[((size_t)bb * NC + C) * SS + s] = sqrtf(sn2);
  }
}

template <int G>
static void run_all(const float* u, float* part, float* warr,
                    float* vout, float* aout, hipStream_t stream) {
  const dim3 accGrid(8 * STILES * G), sqGrid(8 * STILES), blk(256);
  caps_acc_kernel<0, G><<<accGrid, blk, 0, stream>>>(u, warr, part);
  caps_squash_kernel<0, G><<<sqGrid, blk, 0, stream>>>(part, warr, vout, aout);
  caps_acc_kernel<1, G><<<accGrid, blk, 0, stream>>>(u, warr, part);
  caps_squash_kernel<1, G><<<sqGrid, blk, 0, stream>>>(part, warr, vout, aout);
  caps_acc_kernel<2, G><<<accGrid, blk, 0, stream>>>(u, warr, part);
  caps_squash_kernel<2, G><<<sqGrid, blk, 0, stream>>>(part, warr, vout, aout);
}

extern "C" void kernel_launch(void* const* d_in, const int* in_sizes, int n_in,
                              void* d_out, int out_size, void* d_ws, size_t ws_size,
                              hipStream_t stream) {
  const float* u = (const float*)d_in[0];
  // d_in[1] (`a`) provably does not affect the reference outputs (see header).
  float* vout = (float*)d_out;                   // (8,16,16,12,12)
  float* aout = vout + (size_t)PSZ;              // (8,16,12,12)

  // B-split factor from available workspace: need (G+1)*PSZ floats.
  const size_t psz_b = (size_t)PSZ * sizeof(float);
  float* part = (float*)d_ws;
  if (ws_size >= 9 * psz_b) {          // 576 acc blocks, 18 B each
    run_all<8>(u, part, part + (size_t)8 * PSZ, vout, aout, stream);
  } else if (ws_size >= 5 * psz_b) {
    run_all<4>(u, part, part + (size_t)4 * PSZ, vout, aout, stream);
  } else if (ws_size >= 3 * psz_b) {
    run_all<2>(u, part, part + (size_t)2 * PSZ, vout, aout, stream);
  } else {
    run_all<1>(u, part, part + (size_t)1 * PSZ, vout, aout, stream);
  }
}